// GMN_48730698940875
// MI455X (gfx1250) — compile-verified
//
#include <hip/hip_runtime.h>
#include <hip/hip_bf16.h>

typedef __attribute__((ext_vector_type(2))) float v2f;
typedef __attribute__((ext_vector_type(8))) float v8f;

#define ALPHA 0.2f

__device__ __forceinline__ float leaky(float v) { return v >= 0.f ? v : ALPHA * v; }

__device__ __forceinline__ v8f wmma_f32(v2f a, v2f b, v8f c) {
  // V_WMMA_F32_16X16X4_F32 : D = A(16x4) * B(4x16) + C(16x16)
  return __builtin_amdgcn_wmma_f32_16x16x4_f32(false, a, false, b, (short)0, c,
                                               false, false);
}

// ---------------- constants ----------------
// B=64, N=1000, D_IN=256, HID=512, POS=64, D=320, H=4, K=10
// rows = B*N = 64000 ; row tiles = 4000
// output layout (floats): centroids[0,12800) loss[12800] new_adj[12801,19201)
//                         new_feat[19201,224001) h[224001,20704001)

// ---------------- init: copy centroids, zero loss slot ----------------
__global__ void gmn_init(const float* __restrict__ cent, float* __restrict__ out) {
  int i = blockIdx.x * 256 + threadIdx.x;
  if (i < 12800) out[i] = cent[i];
  if (i == 0) out[12800] = 0.f;
}

// ---------------- fused 2-layer MLP -> h[:, 0:256] ----------------
#define LDA 260   // padded stride for 16x256 tile (260%64==4 -> conflict-free)
#define LDX 516   // padded stride for 16x512 tile (516%64==4)
__global__ __launch_bounds__(128) void gmn_mlp(
    const float* __restrict__ xn, const float* __restrict__ w1,
    const float* __restrict__ b1, const float* __restrict__ w2,
    const float* __restrict__ b2, float* __restrict__ h_out) {
  __shared__ float ldsA[16 * LDA];
  __shared__ float ldsX[16 * LDX];
  const int row0 = blockIdx.x * 16;
  const int tid = threadIdx.x;
  const int lane = tid & 31;
  const int wave = tid >> 5;
  const int lm = lane & 15;
  const int lh = lane >> 4;

  for (int i = tid; i < 16 * 256; i += 128) {
    int r = i >> 8, c = i & 255;
    ldsA[r * LDA + c] = xn[(size_t)(row0 + r) * 256 + c];
  }
  __syncthreads();

  // stage 1: out cols 0..511, 32 n-tiles over 4 waves, K=256
  for (int nt = wave; nt < 32; nt += 4) {
    int n0 = nt * 16;
    v8f acc = {};
    for (int k0 = 0; k0 < 256; k0 += 4) {
      int ka = k0 + lh * 2;
      v2f a;
      a.x = ldsA[lm * LDA + ka];
      a.y = ldsA[lm * LDA + ka + 1];
      const float* wp = &w1[(size_t)(n0 + lm) * 256 + ka];
      v2f bf;
      bf.x = wp[0];
      bf.y = wp[1];
      acc = wmma_f32(a, bf, acc);
    }
    int col = n0 + lm;
    float bias = b1[col];
#pragma unroll
    for (int i = 0; i < 8; i++) {
      int m = i + lh * 8;
      ldsX[m * LDX + col] = leaky(acc[i] + bias);
    }
  }
  __syncthreads();

  // stage 2: out cols 0..255, 16 n-tiles over 4 waves, K=512
  for (int nt = wave; nt < 16; nt += 4) {
    int n0 = nt * 16;
    v8f acc = {};
    for (int k0 = 0; k0 < 512; k0 += 4) {
      int ka = k0 + lh * 2;
      v2f a;
      a.x = ldsX[lm * LDX + ka];
      a.y = ldsX[lm * LDX + ka + 1];
      const float* wp = &w2[(size_t)(n0 + lm) * 512 + ka];
      v2f bf;
      bf.x = wp[0];
      bf.y = wp[1];
      acc = wmma_f32(a, bf, acc);
    }
    int col = n0 + lm;
    float bias = b2[col];
#pragma unroll
    for (int i = 0; i < 8; i++) {
      int m = i + lh * 8;
      h_out[(size_t)(row0 + m) * 320 + col] = leaky(acc[i] + bias);
    }
  }
}

// ---------------- adj projection -> h[:, 256:320] ----------------
#define LDJ 1004  // padded stride (1004%64==44, gcd 4 -> 16 distinct banks)
__global__ __launch_bounds__(128) void gmn_af(
    const float* __restrict__ adj, const float* __restrict__ wa,
    const float* __restrict__ ba, float* __restrict__ h_out) {
  __shared__ float ldsA[16 * LDJ];
  const int row0 = blockIdx.x * 16;
  const int tid = threadIdx.x;
  const int lane = tid & 31;
  const int wave = tid >> 5;
  const int lm = lane & 15;
  const int lh = lane >> 4;

  for (int i = tid; i < 16 * 1000; i += 128) {
    int r = i / 1000, c = i - r * 1000;
    ldsA[r * LDJ + c] = adj[(size_t)row0 * 1000 + i];
  }
  __syncthreads();

  int n0 = wave * 16;  // 4 tiles of 16 = POS 64
  v8f acc = {};
  for (int k0 = 0; k0 < 1000; k0 += 4) {
    int ka = k0 + lh * 2;
    v2f a;
    a.x = ldsA[lm * LDJ + ka];
    a.y = ldsA[lm * LDJ + ka + 1];
    const float* wp = &wa[(size_t)(n0 + lm) * 1000 + ka];
    v2f bf;
    bf.x = wp[0];
    bf.y = wp[1];
    acc = wmma_f32(a, bf, acc);
  }
  int col = n0 + lm;
  float bias = ba[col];
#pragma unroll
  for (int i = 0; i < 8; i++) {
    int m = i + lh * 8;
    h_out[(size_t)(row0 + m) * 320 + 256 + col] = leaky(acc[i] + bias);
  }
}

// ---------------- per-node q (dist, per-head norm, mean, softmax) ----------
__global__ __launch_bounds__(256) void gmn_q(
    const float* __restrict__ hbuf, const float* __restrict__ cent,
    const int* __restrict__ gsz, float* __restrict__ q) {
  __shared__ float lc[40 * 320];
  __shared__ float lc2[40];
  int tid = threadIdx.x;
  for (int i = tid; i < 40 * 320; i += 256) lc[i] = cent[i];
  __syncthreads();
  if (tid < 40) {
    float s = 0.f;
    for (int d = 0; d < 320; d++) {
      float v = lc[tid * 320 + d];
      s += v * v;
    }
    lc2[tid] = s;
  }
  __syncthreads();

  int row = blockIdx.x * 256 + tid;
  if (row >= 64000) return;
  int b = row / 1000, n = row - b * 1000;
  const float* hv = &hbuf[(size_t)row * 320];

  float xc[40];
#pragma unroll
  for (int j = 0; j < 40; j++) xc[j] = 0.f;
  float x2 = 0.f;
  for (int d = 0; d < 320; d++) {
    float hd = hv[d];
    x2 += hd * hd;
#pragma unroll
    for (int j = 0; j < 40; j++) xc[j] += hd * lc[j * 320 + d];
  }

  float mf = (n < gsz[b]) ? 1.f : 0.f;
  float qm[10];
#pragma unroll
  for (int k = 0; k < 10; k++) qm[k] = 0.f;
  for (int hh = 0; hh < 4; hh++) {
    float qv[10];
    float s = 0.f;
#pragma unroll
    for (int k = 0; k < 10; k++) {
      int j = hh * 10 + k;
      float dist = fmaxf(x2 + lc2[j] - 2.f * xc[j], 0.f) * mf;
      float qq = 1.f / (1.f + dist);
      qv[k] = qq;
      s += qq;
    }
#pragma unroll
    for (int k = 0; k < 10; k++) qm[k] += (qv[k] / s) * mf;
  }
  float mx = -1e30f;
#pragma unroll
  for (int k = 0; k < 10; k++) {
    qm[k] *= 0.25f;
    mx = fmaxf(mx, qm[k]);
  }
  float se = 0.f;
#pragma unroll
  for (int k = 0; k < 10; k++) {
    qm[k] = expf(qm[k] - mx);
    se += qm[k];
  }
#pragma unroll
  for (int k = 0; k < 10; k++)
    q[(size_t)b * 10000 + k * 1000 + n] = (qm[k] / se) * mf;
}

// ---------------- column sums of q over n ----------------
__global__ void gmn_colsum(const float* __restrict__ q, float* __restrict__ cs) {
  int idx = blockIdx.x * 128 + threadIdx.x;
  if (idx >= 640) return;
  const float* p = &q[(size_t)idx * 1000];
  float s = 0.f;
  for (int n = 0; n < 1000; n++) s += p[n];
  cs[idx] = s;
}

// ---------------- KL hard loss ----------------
__global__ __launch_bounds__(256) void gmn_loss(
    const float* __restrict__ q, const float* __restrict__ cs,
    const int* __restrict__ gsz, float* __restrict__ loss_out) {
  __shared__ float red[256];
  int tid = threadIdx.x;
  int row = blockIdx.x * 256 + tid;
  float local = 0.f;
  if (row < 64000) {
    int b = row / 1000, n = row - b * 1000;
    if (n < gsz[b]) {
      float qv[10], pv[10];
      float den = 0.f;
#pragma unroll
      for (int k = 0; k < 10; k++) {
        float qq = q[(size_t)(b * 10 + k) * 1000 + n];
        float pp = qq * qq / cs[b * 10 + k];
        qv[k] = qq;
        pv[k] = pp;
        den += pp;
      }
#pragma unroll
      for (int k = 0; k < 10; k++) {
        float pp = pv[k] / den;
        local += pp * logf(pp / qv[k]);
      }
    }
  }
  red[tid] = local;
  __syncthreads();
  for (int s = 128; s > 0; s >>= 1) {
    if (tid < s) red[tid] += red[tid + s];
    __syncthreads();
  }
  if (tid == 0) atomicAdd(loss_out, 100.f * red[0]);
}

// ---------------- q_adj = q @ adj  (per b: 10x1000 = (10x1000)(1000x1000)) --
__global__ __launch_bounds__(128) void gmn_qadj(
    const float* __restrict__ q, const float* __restrict__ adj,
    float* __restrict__ qa) {
  int wave = (blockIdx.x * 128 + threadIdx.x) >> 5;
  int lane = threadIdx.x & 31;
  if (wave >= 64 * 63) return;
  int b = wave / 63;
  int nt = wave - b * 63;
  int n0 = nt * 16;
  int lm = lane & 15, lh = lane >> 4;
  const float* qb = &q[(size_t)b * 10000];
  const float* ab = &adj[(size_t)b * 1000000];
  bool arow = lm < 10;
  int ncol = n0 + lm;
  bool cvalid = ncol < 1000;
  v8f acc = {};
  for (int k0 = 0; k0 < 1000; k0 += 4) {
    int ka = k0 + lh * 2;
    v2f a;
    a.x = arow ? qb[lm * 1000 + ka] : 0.f;
    a.y = arow ? qb[lm * 1000 + ka + 1] : 0.f;
    v2f bf;
    bf.x = cvalid ? ab[(size_t)ka * 1000 + ncol] : 0.f;
    bf.y = cvalid ? ab[(size_t)(ka + 1) * 1000 + ncol] : 0.f;
    acc = wmma_f32(a, bf, acc);
  }
  if (cvalid) {
#pragma unroll
    for (int i = 0; i < 8; i++) {
      int m = i + lh * 8;
      if (m < 10) qa[(size_t)b * 10000 + m * 1000 + ncol] = acc[i];
    }
  }
}

// ---------------- new_adj = q_adj @ q^T (per b: 10x10) ----------------
__global__ __launch_bounds__(128) void gmn_newadj(
    const float* __restrict__ q, const float* __restrict__ qa,
    float* __restrict__ out) {
  int wave = (blockIdx.x * 128 + threadIdx.x) >> 5;
  int lane = threadIdx.x & 31;
  if (wave >= 64) return;
  int b = wave;
  int lm = lane & 15, lh = lane >> 4;
  const float* qb = &q[(size_t)b * 10000];
  const float* qab = &qa[(size_t)b * 10000];
  bool valid = lm < 10;
  v8f acc = {};
  for (int k0 = 0; k0 < 1000; k0 += 4) {
    int ka = k0 + lh * 2;
    v2f a;
    a.x = valid ? qab[lm * 1000 + ka] : 0.f;
    a.y = valid ? qab[lm * 1000 + ka + 1] : 0.f;
    v2f bf;  // B[k][j] = q[b][j][k]
    bf.x = valid ? qb[lm * 1000 + ka] : 0.f;
    bf.y = valid ? qb[lm * 1000 + ka + 1] : 0.f;
    acc = wmma_f32(a, bf, acc);
  }
#pragma unroll
  for (int i = 0; i < 8; i++) {
    int m = i + lh * 8;
    if (m < 10 && lm < 10) out[b * 100 + m * 10 + lm] = acc[i];
  }
}

// ---------------- new_feat = q @ h (per b: 10x320) ----------------
__global__ __launch_bounds__(128) void gmn_newfeat(
    const float* __restrict__ q, const float* __restrict__ hbuf,
    float* __restrict__ out) {
  int wave = (blockIdx.x * 128 + threadIdx.x) >> 5;
  int lane = threadIdx.x & 31;
  if (wave >= 64 * 20) return;
  int b = wave / 20;
  int dt = wave - b * 20;
  int d0 = dt * 16;
  int lm = lane & 15, lh = lane >> 4;
  const float* qb = &q[(size_t)b * 10000];
  const float* hb = &hbuf[(size_t)b * 1000 * 320];
  bool arow = lm < 10;
  v8f acc = {};
  for (int k0 = 0; k0 < 1000; k0 += 4) {
    int ka = k0 + lh * 2;
    v2f a;
    a.x = arow ? qb[lm * 1000 + ka] : 0.f;
    a.y = arow ? qb[lm * 1000 + ka + 1] : 0.f;
    v2f bf;  // B[k][n] = h[b][k][d0+n]
    bf.x = hb[(size_t)ka * 320 + d0 + lm];
    bf.y = hb[(size_t)(ka + 1) * 320 + d0 + lm];
    acc = wmma_f32(a, bf, acc);
  }
#pragma unroll
  for (int i = 0; i < 8; i++) {
    int m = i + lh * 8;
    if (m < 10) out[(size_t)b * 3200 + m * 320 + d0 + lm] = acc[i];
  }
}

extern "C" void kernel_launch(void* const* d_in, const int* in_sizes, int n_in,
                              void* d_out, int out_size, void* d_ws,
                              size_t ws_size, hipStream_t stream) {
  const float* x_node = (const float*)d_in[0];
  const float* adj    = (const float*)d_in[1];
  const float* w1     = (const float*)d_in[2];
  const float* b1     = (const float*)d_in[3];
  const float* w2     = (const float*)d_in[4];
  const float* b2     = (const float*)d_in[5];
  const float* wa     = (const float*)d_in[6];
  const float* ba     = (const float*)d_in[7];
  const float* cent   = (const float*)d_in[8];
  const int*   gsz    = (const int*)d_in[9];

  float* out = (float*)d_out;
  // output offsets (floats)
  float* out_cent    = out;              // 12800
  float* out_loss    = out + 12800;      // 1
  float* out_newadj  = out + 12801;      // 6400
  float* out_newfeat = out + 19201;      // 204800
  float* out_h       = out + 224001;     // 64000*320

  // workspace: q (640000) | colsum (640) | q_adj (640000)
  float* ws_q  = (float*)d_ws;
  float* ws_cs = ws_q + 640000;
  float* ws_qa = ws_cs + 640;

  gmn_init<<<50, 256, 0, stream>>>(cent, out);
  gmn_mlp<<<4000, 128, 0, stream>>>(x_node, w1, b1, w2, b2, out_h);
  gmn_af<<<4000, 128, 0, stream>>>(adj, wa, ba, out_h);
  gmn_q<<<250, 256, 0, stream>>>(out_h, cent, gsz, ws_q);
  gmn_colsum<<<5, 128, 0, stream>>>(ws_q, ws_cs);
  gmn_loss<<<250, 256, 0, stream>>>(ws_q, ws_cs, gsz, out_loss);
  gmn_qadj<<<1008, 128, 0, stream>>>(ws_q, adj, ws_qa);
  gmn_newadj<<<16, 128, 0, stream>>>(ws_q, ws_qa, out_newadj);
  gmn_newfeat<<<320, 128, 0, stream>>>(ws_q, out_h, out_newfeat);
}